// TimeAwareMultiHeadAttention_20959440404898
// MI455X (gfx1250) — compile-verified
//
#include <hip/hip_runtime.h>
#include <hip/hip_bf16.h>
#include <math.h>

#define BATCH 16
#define SEQ   200
#define HID   256
#define NHEAD 4
#define HD    64
#define NEGV  (-4294967295.0f)   // -(2^32 - 1), matches reference padding value

typedef __attribute__((ext_vector_type(2))) float v2f;
typedef __attribute__((ext_vector_type(4))) float v4f;
typedef __attribute__((ext_vector_type(8))) float v8f;

// ---------------------------------------------------------------------------
// Kernel 1: fused linear projections with V_WMMA_F32_16X16X4_F32 (fp32 WMMA).
//   sel=0: Qb   = queries @ Wq^T + bq
//   sel=1: Keff = keys    @ Wk^T + bk + abs_pos_K   (posK folded into K)
//   sel=2: Veff = keys    @ Wv^T + bv + abs_pos_V   (posV folded into V)
// One wave per 16x16 output tile. A-matrix 16x4 layout: lane = M + 16*(K>>1),
// vgpr = K&1 -> lane loads a contiguous float2 at column k + 2*(lane>>4).
// B-matrix 4x16 mirrors it with N in place of M. EXEC all-ones (blockDim=32).
// ---------------------------------------------------------------------------
__global__ __launch_bounds__(32) void proj_wmma_kernel(
    const float* __restrict__ queries, const float* __restrict__ keys,
    const float* __restrict__ Wq, const float* __restrict__ bq,
    const float* __restrict__ Wk, const float* __restrict__ bk,
    const float* __restrict__ Wv, const float* __restrict__ bv,
    const float* __restrict__ posK, const float* __restrict__ posV,
    float* __restrict__ Qb, float* __restrict__ Keff, float* __restrict__ Veff)
{
    const int lane = threadIdx.x;          // 0..31 (wave32)
    const int r0   = blockIdx.x * 16;      // token-row tile (B*L = 3200 = 200*16)
    const int j0   = blockIdx.y * 16;      // output-channel tile
    const int sel  = blockIdx.z;

    const float* In   = (sel == 0) ? queries : keys;
    const float* W    = (sel == 0) ? Wq : (sel == 1) ? Wk : Wv;
    const float* bias = (sel == 0) ? bq : (sel == 1) ? bk : bv;

    const int M  = lane & 15;
    const int kh = (lane >> 4) << 1;       // 0 or 2

    const float* arow = In + (size_t)(r0 + M) * HID + kh;
    const float* brow = W  + (size_t)(j0 + M) * HID + kh;   // N == M formula

    v8f acc = {0.f, 0.f, 0.f, 0.f, 0.f, 0.f, 0.f, 0.f};
    #pragma unroll 8
    for (int k = 0; k < HID; k += 4) {
        v2f a  = *(const v2f*)(arow + k);
        v2f bb = *(const v2f*)(brow + k);
        // (neg_a, A, neg_b, B, c_mod, C, reuse_a, reuse_b)
        acc = __builtin_amdgcn_wmma_f32_16x16x4_f32(false, a, false, bb,
                                                    (short)0, acc, false, false);
    }

    float*       Out = (sel == 0) ? Qb : (sel == 1) ? Keff : Veff;
    const float* add = (sel == 1) ? posK : posV;
    const int col   = j0 + M;
    const float bj  = bias[col];
    const int rbase = r0 + ((lane >> 4) << 3);   // C layout: M = v + 8*(lane/16)
    #pragma unroll
    for (int v = 0; v < 8; ++v) {
        const int r = rbase + v;
        float val = acc[v] + bj;
        if (sel != 0) val += add[(size_t)r * HID + col];
        Out[(size_t)r * HID + col] = val;
    }
}

// ---------------------------------------------------------------------------
// Kernel 2: streaming attention, one 256-thread block (8 wave32) per (b,l).
// Phase 1: scores[h,m] = (Q . (Keff_m + tmK[b,l,m])) / 8, causal+row mask.
// Phase 2: per-head softmax (wave32 shfl_xor reduce + LDS pair combine).
// Phase 3: out[c] = sum_m A[h(c),m] * (Veff[m,c] + tmV[b,l,m,c]), coalesced.
// ---------------------------------------------------------------------------
__global__ __launch_bounds__(256) void attn_stream_kernel(
    const float* __restrict__ Qb, const float* __restrict__ Keff,
    const float* __restrict__ Veff,
    const float* __restrict__ tmK, const float* __restrict__ tmV,
    const unsigned char* __restrict__ time_mask,
    float* __restrict__ out)
{
    __shared__ float s_q[HID];
    __shared__ float s_a[NHEAD * SEQ];
    __shared__ float s_red_a[8];
    __shared__ float s_red_b[8];

    const int tid = threadIdx.x;
    const int bl  = blockIdx.x;            // b*SEQ + l
    const int b   = bl / SEQ;
    const int l   = bl - b * SEQ;
    const bool rowmask = (time_mask[bl] != 0);

    s_q[tid] = Qb[(size_t)bl * HID + tid];
    __syncthreads();

    // ---- Phase 1: masked scores ----
    const float  scale    = 0.125f;                      // 1/sqrt(64)
    const size_t tmk_base = (size_t)bl * SEQ * HID;
    for (int p = tid; p < NHEAD * SEQ; p += 256) {
        const int h = p / SEQ;
        const int m = p - h * SEQ;
        float sc;
        if (rowmask || m > l) {
            sc = NEGV;                                    // skip loads entirely
        } else {
            const float* kp = Keff + ((size_t)(b * SEQ + m)) * HID + h * HD;
            const float* tp = tmK + tmk_base + (size_t)m * HID + h * HD;
            const float* qp = s_q + h * HD;
            float acc = 0.f;
            #pragma unroll 4
            for (int d = 0; d < HD; d += 4) {
                v4f kv = *(const v4f*)(kp + d);
                v4f tv = *(const v4f*)(tp + d);
                acc = fmaf(qp[d + 0], kv.x + tv.x, acc);
                acc = fmaf(qp[d + 1], kv.y + tv.y, acc);
                acc = fmaf(qp[d + 2], kv.z + tv.z, acc);
                acc = fmaf(qp[d + 3], kv.w + tv.w, acc);
            }
            sc = acc * scale;
        }
        s_a[p] = sc;
    }
    __syncthreads();

    // ---- Phase 2: per-head softmax over m ----
    {
        const int h  = tid >> 6;                          // 64 threads per head
        const int i  = tid & 63;
        const int wv = tid >> 5;

        float pmax = NEGV;
        for (int m = i; m < SEQ; m += 64) pmax = fmaxf(pmax, s_a[h * SEQ + m]);
        #pragma unroll
        for (int off = 16; off > 0; off >>= 1)
            pmax = fmaxf(pmax, __shfl_xor(pmax, off, 32));
        if ((tid & 31) == 0) s_red_a[wv] = pmax;
        __syncthreads();
        const float mx = fmaxf(s_red_a[2 * h], s_red_a[2 * h + 1]);

        float psum = 0.f;
        for (int m = i; m < SEQ; m += 64) {
            const float e = __expf(s_a[h * SEQ + m] - mx);  // NEG-mx -> 0 exactly
            s_a[h * SEQ + m] = e;
            psum += e;
        }
        #pragma unroll
        for (int off = 16; off > 0; off >>= 1)
            psum += __shfl_xor(psum, off, 32);
        if ((tid & 31) == 0) s_red_b[wv] = psum;
        __syncthreads();
        const float inv = 1.f / (s_red_b[2 * h] + s_red_b[2 * h + 1]);
        for (int m = i; m < SEQ; m += 64) s_a[h * SEQ + m] *= inv;
        __syncthreads();
    }

    // ---- Phase 3: output accumulation, channel-per-thread (fully coalesced) ----
    {
        const int c  = tid;
        const int h  = c >> 6;
        const float* arow = s_a + h * SEQ;
        const size_t tmv_base = (size_t)bl * SEQ * HID;
        float acc = 0.f;
        for (int m = 0; m < SEQ; ++m) {
            const float a = arow[m];                     // LDS broadcast in wave
            if (a != 0.f) {                              // uniform branch per wave
                const float ve = Veff[((size_t)(b * SEQ + m)) * HID + c];
                const float tv = tmV[tmv_base + (size_t)m * HID + c];
                acc = fmaf(a, ve + tv, acc);
            }
        }
        out[(size_t)bl * HID + c] = acc;
    }
}

// ---------------------------------------------------------------------------
extern "C" void kernel_launch(void* const* d_in, const int* in_sizes, int n_in,
                              void* d_out, int out_size, void* d_ws, size_t ws_size,
                              hipStream_t stream) {
    const float* queries = (const float*)d_in[0];
    const float* keys    = (const float*)d_in[1];
    const unsigned char* time_mask = (const unsigned char*)d_in[2];
    // d_in[3] attn_mask: causal triu(k=1) -> computed analytically (m > l)
    const float* tmK  = (const float*)d_in[4];
    const float* tmV  = (const float*)d_in[5];
    const float* posK = (const float*)d_in[6];
    const float* posV = (const float*)d_in[7];
    const float* Wq = (const float*)d_in[8];
    const float* bq = (const float*)d_in[9];
    const float* Wk = (const float*)d_in[10];
    const float* bk = (const float*)d_in[11];
    const float* Wv = (const float*)d_in[12];
    const float* bv = (const float*)d_in[13];
    float* out = (float*)d_out;

    float* Qb   = (float*)d_ws;                 // 3 x 3200 x 256 fp32 = 9.8 MB
    float* Keff = Qb   + (size_t)BATCH * SEQ * HID;
    float* Veff = Keff + (size_t)BATCH * SEQ * HID;

    dim3 gp(BATCH * SEQ / 16, HID / 16, 3);     // 200 x 16 x 3 tiles
    proj_wmma_kernel<<<gp, 32, 0, stream>>>(queries, keys, Wq, bq, Wk, bk,
                                            Wv, bv, posK, posV, Qb, Keff, Veff);

    attn_stream_kernel<<<BATCH * SEQ, 256, 0, stream>>>(Qb, Keff, Veff,
                                                        tmK, tmV, time_mask, out);
}